// GCNClassifier_51453708206833
// MI455X (gfx1250) — compile-verified
//
#include <hip/hip_runtime.h>
#include <hip/hip_bf16.h>

typedef __attribute__((ext_vector_type(2))) float v2f;
typedef __attribute__((ext_vector_type(8))) float v8f;

#define DH 64

// ---------------------------------------------------------------------------
// Utility: fill a float buffer with a constant
// ---------------------------------------------------------------------------
__global__ void fill_kernel(float* __restrict__ p, float v, int n) {
  int t = blockIdx.x * blockDim.x + threadIdx.x;
  if (t < n) p[t] = v;
}

// ---------------------------------------------------------------------------
// Degree accumulation: deg[dst[e]] += 1 (deg pre-initialized to 1.0 = self loop)
// ---------------------------------------------------------------------------
__global__ void deg_kernel(const int* __restrict__ dst, float* __restrict__ deg, int E) {
  int e = blockIdx.x * blockDim.x + threadIdx.x;
  if (e < E) atomicAdd(&deg[dst[e]], 1.0f);
}

// In-place deg -> rsqrt(deg). deg >= 1 always (self loops), so no zero guard.
__global__ void rsqrt_kernel(float* __restrict__ deg, int N) {
  int n = blockIdx.x * blockDim.x + threadIdx.x;
  if (n < N) deg[n] = rsqrtf(deg[n]);
}

// ---------------------------------------------------------------------------
// WMMA f32 GEMM:  H[N x 64] = X[N x K] @ W[K x 64]   (K = 128 or 64)
// Block = 128 threads = 4 waves. Block covers 16 rows x 64 cols.
// Wave w computes the 16x16 tile at cols [16w, 16w+16) using
// V_WMMA_F32_16X16X4_F32, accumulating over K in steps of 4.
//
// Fragment layouts (CDNA5 ISA 7.12.2, wave32):
//   A 16x4 (MxK):  lane<16: M=lane,  {a.x,a.y} = K {k,k+1}
//                  lane>=16: M=lane-16, {a.x,a.y} = K {k+2,k+3}
//   B 4x16 (KxN):  row striped across lanes:
//                  lane<16: N=lane,  {b.x,b.y} = K {k,k+1}
//                  lane>=16: N=lane-16, {b.x,b.y} = K {k+2,k+3}
//   C/D 16x16:     VGPR r, lane<16: (M=r, N=lane); lane>=16: (M=r+8, N=lane-16)
// ---------------------------------------------------------------------------
template <int K>
__global__ void gemm_wmma_f32(const float* __restrict__ X,
                              const float* __restrict__ W,
                              float* __restrict__ H, int N) {
  __shared__ float lW[K * DH];           // 32KB (K=128) or 16KB (K=64) of 320KB WGP LDS
  int tid = threadIdx.x;                  // 0..127
  for (int i = tid; i < K * DH; i += 128) lW[i] = W[i];
  __syncthreads();

  const int wave = tid >> 5;              // 0..3 -> column tile
  const int lane = tid & 31;
  const int mr = lane & 15;               // A row / B-C column within tile
  const int hi = lane >> 4;               // K-pair selector
  const int rowBase = blockIdx.x * 16;
  const int colBase = wave * 16;

  // clamp row so loads stay in range even for ragged N; stores are guarded
  const int arow = (rowBase + mr < N) ? (rowBase + mr) : (N - 1);
  const float* __restrict__ xrow = X + (size_t)arow * K;

  v8f c = {0.f, 0.f, 0.f, 0.f, 0.f, 0.f, 0.f, 0.f};
#pragma unroll
  for (int k = 0; k < K; k += 4) {
    const int kk = k + hi * 2;
    v2f a, b;
    a.x = xrow[kk];
    a.y = xrow[kk + 1];
    b.x = lW[kk * DH + colBase + mr];
    b.y = lW[(kk + 1) * DH + colBase + mr];
    // (neg_a, A, neg_b, B, c_mod, C, reuse_a, reuse_b)
    c = __builtin_amdgcn_wmma_f32_16x16x4_f32(false, a, false, b, (short)0, c,
                                              false, false);
  }

#pragma unroll
  for (int r = 0; r < 8; ++r) {
    int row = rowBase + r + hi * 8;
    if (row < N) H[(size_t)row * DH + colBase + mr] = c[r];
  }
}

// ---------------------------------------------------------------------------
// Edge message scatter: accum[dst] += h[src] * dinv[src]*dinv[dst]
// 32 lanes per edge, float2 per lane covers all 64 features.
// ---------------------------------------------------------------------------
__global__ void edge_scatter(const int* __restrict__ src, const int* __restrict__ dst,
                             const float* __restrict__ dinv, const float* __restrict__ h,
                             float* __restrict__ accum, int E) {
  int t = blockIdx.x * blockDim.x + threadIdx.x;
  int e = t >> 5;
  if (e >= E) return;
  int c2 = (t & 31) * 2;
  int s = src[e];
  int d = dst[e];
  float w = dinv[s] * dinv[d];
  const float2 hv = *reinterpret_cast<const float2*>(h + (size_t)s * DH + c2);
  atomicAdd(&accum[(size_t)d * DH + c2],     hv.x * w);
  atomicAdd(&accum[(size_t)d * DH + c2 + 1], hv.y * w);
}

// ---------------------------------------------------------------------------
// Epilogue: fold in self-loop message (h[n]*dinv[n]^2), bias, ReLU. In place.
// ---------------------------------------------------------------------------
__global__ void bias_relu(float* __restrict__ a, const float* __restrict__ h,
                          const float* __restrict__ dinv, const float* __restrict__ b,
                          int total) {
  int t = blockIdx.x * blockDim.x + threadIdx.x;
  if (t >= total) return;
  int n = t >> 6;
  int d = t & 63;
  float di = dinv[n];
  float v = a[t] + h[t] * di * di + b[d];
  a[t] = v > 0.f ? v : 0.f;
}

// ---------------------------------------------------------------------------
// Mean-pool sums over sorted batch: run-length accumulate, flush on graph change.
// 256 threads: 128 feature columns x 2 node phases; 256 nodes per block.
// ---------------------------------------------------------------------------
__global__ void pool_kernel(const int* __restrict__ batch, const float* __restrict__ x1,
                            const float* __restrict__ x2, float* __restrict__ pool, int N) {
  const int NPB = 256;
  int d = threadIdx.x & 127;
  int half = threadIdx.x >> 7;
  int n0 = blockIdx.x * NPB;
  int nEnd = n0 + NPB < N ? n0 + NPB : N;
  float acc = 0.f;
  int gcur = -1;
  for (int n = n0 + half; n < nEnd; n += 2) {
    int g = batch[n];
    float v = (d < 64) ? x1[(size_t)n * DH + d] : x2[(size_t)n * DH + (d - 64)];
    if (g != gcur) {
      if (gcur >= 0) atomicAdd(&pool[gcur * 128 + d], acc);
      gcur = g;
      acc = v;
    } else {
      acc += v;
    }
  }
  if (gcur >= 0) atomicAdd(&pool[gcur * 128 + d], acc);
}

// Per-graph node counts with LDS pre-reduction (64 graphs).
__global__ void count_kernel(const int* __restrict__ batch, float* __restrict__ cnt, int N) {
  __shared__ float lc[64];
  if (threadIdx.x < 64) lc[threadIdx.x] = 0.f;
  __syncthreads();
  int n = blockIdx.x * blockDim.x + threadIdx.x;
  if (n < N) atomicAdd(&lc[batch[n]], 1.0f);
  __syncthreads();
  if (threadIdx.x < 64 && lc[threadIdx.x] != 0.f)
    atomicAdd(&cnt[threadIdx.x], lc[threadIdx.x]);
}

// ---------------------------------------------------------------------------
// Head: out[64 x 7] = (pool/count) @ Wlin + blin. Tiny; one block.
// ---------------------------------------------------------------------------
__global__ void final_linear(const float* __restrict__ pool, const float* __restrict__ cnt,
                             const float* __restrict__ Wlin, const float* __restrict__ blin,
                             float* __restrict__ out) {
  int t = threadIdx.x;
  if (t >= 64 * 7) return;
  int g = t / 7;
  int o = t % 7;
  float inv = 1.0f / fmaxf(cnt[g], 1.0f);
  float s = blin[o];
#pragma unroll 8
  for (int k = 0; k < 128; ++k) s += pool[g * 128 + k] * inv * Wlin[k * 7 + o];
  out[t] = s;
}

// ---------------------------------------------------------------------------
extern "C" void kernel_launch(void* const* d_in, const int* in_sizes, int n_in,
                              void* d_out, int out_size, void* d_ws, size_t ws_size,
                              hipStream_t stream) {
  const float* x    = (const float*)d_in[0];
  const int*   ei   = (const int*)d_in[1];   // int32 (JAX x64 disabled)
  const int*   bat  = (const int*)d_in[2];
  const float* W1   = (const float*)d_in[3];
  const float* b1   = (const float*)d_in[4];
  const float* W2   = (const float*)d_in[5];
  const float* b2   = (const float*)d_in[6];
  const float* Wlin = (const float*)d_in[7];
  const float* blin = (const float*)d_in[8];
  float* out = (float*)d_out;

  const int N = in_sizes[0] / 128;   // 50000
  const int E = in_sizes[1] / 2;     // 800000
  const int* src = ei;
  const int* dst = ei + E;

  // workspace carve-up (floats)
  float* ws   = (float*)d_ws;
  float* dinv = ws;                          // N
  float* h1   = dinv + N;                    // N*64
  float* a1   = h1 + (size_t)N * DH;         // N*64 (accum -> x1)
  float* h2   = a1 + (size_t)N * DH;         // N*64
  float* a2   = h2 + (size_t)N * DH;         // N*64 (accum -> x2)
  float* pool = a2 + (size_t)N * DH;         // 64*128
  float* cnt  = pool + 64 * 128;             // 64

  const int TPB = 256;
  auto blks = [](long long n, int t) { return (unsigned)((n + t - 1) / t); };

  // --- init ---
  fill_kernel<<<blks(N, TPB), TPB, 0, stream>>>(dinv, 1.0f, N);            // self loop
  fill_kernel<<<blks((long long)N * DH, TPB), TPB, 0, stream>>>(a1, 0.f, N * DH);
  fill_kernel<<<blks((long long)N * DH, TPB), TPB, 0, stream>>>(a2, 0.f, N * DH);
  fill_kernel<<<blks(64 * 128 + 64, TPB), TPB, 0, stream>>>(pool, 0.f, 64 * 128 + 64);

  // --- normalization coefficients ---
  deg_kernel<<<blks(E, TPB), TPB, 0, stream>>>(dst, dinv, E);
  rsqrt_kernel<<<blks(N, TPB), TPB, 0, stream>>>(dinv, N);

  // --- conv 1: h1 = x @ W1 (WMMA), scatter, epilogue -> x1 (in a1) ---
  gemm_wmma_f32<128><<<blks(N, 16), 128, 0, stream>>>(x, W1, h1, N);
  edge_scatter<<<blks((long long)E * 32, TPB), TPB, 0, stream>>>(src, dst, dinv, h1, a1, E);
  bias_relu<<<blks((long long)N * DH, TPB), TPB, 0, stream>>>(a1, h1, dinv, b1, N * DH);

  // --- conv 2: h2 = x1 @ W2 (WMMA), scatter, epilogue -> x2 (in a2) ---
  gemm_wmma_f32<64><<<blks(N, 16), 128, 0, stream>>>(a1, W2, h2, N);
  edge_scatter<<<blks((long long)E * 32, TPB), TPB, 0, stream>>>(src, dst, dinv, h2, a2, E);
  bias_relu<<<blks((long long)N * DH, TPB), TPB, 0, stream>>>(a2, h2, dinv, b2, N * DH);

  // --- JK-cat mean pool + head ---
  pool_kernel<<<blks(N, 256), 256, 0, stream>>>(bat, a1, a2, pool, N);
  count_kernel<<<blks(N, 256), 256, 0, stream>>>(bat, cnt, N);
  final_linear<<<1, 512, 0, stream>>>(pool, cnt, Wlin, blin, out);
}